// KPInvBlock_66271345377641
// MI455X (gfx1250) — compile-verified
//
#include <hip/hip_runtime.h>
#include <hip/hip_bf16.h>
#include <math.h>

typedef float v2f __attribute__((ext_vector_type(2)));
typedef float v8f __attribute__((ext_vector_type(8)));

#define CC        128     // channels
#define KK        15      // kernel points (padded to 16)
#define HH        32      // neighbors
#define HID       32      // hidden dim of MLP
#define KGDIM     120     // K*G
#define GG        8       // groups
#define WAVES     4       // waves per block
#define FSTRIDE   132     // feats LDS row stride (dwords), keeps 16B align, spreads banks
#define ISTRIDE   33      // influence LDS row stride (dwords)
#define INV_SIGMA 0.5f    // 1/SIGMA, SIGMA=2.0
#define SLOPE     0.1f

// ---------------------------------------------------------------------------
// Main kernel: one wave32 per query point.
//  - gather neighbor feats -> LDS (row per neighbor)
//  - influence (16x32, row 15 zero) -> LDS
//  - tiny MLP (128->32->120) -> conv weights in LDS (padded to 128, k=15 zero)
//  - einsum via V_WMMA_F32_16X16X4_F32: 8 c-tiles x 8 h-chunks
//  - K-reduction with conv weights, write pre-norm rows to `out` (= d_out)
// ---------------------------------------------------------------------------
__global__ __launch_bounds__(WAVES * 32) void kp_main(
    const float* __restrict__ q_pts,    // (M,3)
    const float* __restrict__ s_pts,    // (N,3)
    const float* __restrict__ s_feats,  // (N,128)
    const int*   __restrict__ nbr,      // (M,32)
    const float* __restrict__ kpts,     // (15,3)
    const float* __restrict__ W1,       // (128,32)
    const float* __restrict__ b1,       // (32)
    const float* __restrict__ W2,       // (32,120)
    const float* __restrict__ b2,       // (120)
    float* __restrict__ out,            // (M,128) pre-norm
    int M)
{
  __shared__ float Fs[WAVES][HH * FSTRIDE];   // gathered feats, per wave
  __shared__ float Is[WAVES][16 * ISTRIDE];   // influence 16x32 (row 15 = 0)
  __shared__ float H1[WAVES][HID];            // MLP hidden
  __shared__ float CW[WAVES][128];            // conv weights, j=k*8+g, [120..127]=0

  const int lane  = threadIdx.x & 31;
  const int w     = threadIdx.x >> 5;
  const int m0    = blockIdx.x * WAVES + w;
  const bool act  = (m0 < M);
  const int m     = act ? m0 : 0;

  float* F = Fs[w];
  float* I = Is[w];

  // ---- per-lane neighbor ----
  const int ni = nbr[m * HH + lane];

  const float qx = q_pts[m * 3 + 0];
  const float qy = q_pts[m * 3 + 1];
  const float qz = q_pts[m * 3 + 2];
  const float rx = s_pts[(size_t)ni * 3 + 0] - qx;
  const float ry = s_pts[(size_t)ni * 3 + 1] - qy;
  const float rz = s_pts[(size_t)ni * 3 + 2] - qz;

  // ---- influence: lane = neighbor h, loop kernel points ----
  #pragma unroll
  for (int k = 0; k < KK; ++k) {
    const float dx = rx - kpts[k * 3 + 0];
    const float dy = ry - kpts[k * 3 + 1];
    const float dz = rz - kpts[k * 3 + 2];
    const float d2 = dx * dx + dy * dy + dz * dz;
    I[k * ISTRIDE + lane] = fmaxf(1.0f - sqrtf(d2) * INV_SIGMA, 0.0f);
  }
  I[15 * ISTRIDE + lane] = 0.0f;  // pad row

  // ---- gather feats row (128 f32) for this lane's neighbor into LDS ----
  {
    const float4* src = (const float4*)(s_feats + (size_t)ni * CC);
    float4* dst = (float4*)(F + lane * FSTRIDE);
    #pragma unroll 8
    for (int j = 0; j < CC / 4; ++j) dst[j] = src[j];
  }
  __syncthreads();

  // ---- MLP hidden: lane = hidden unit, center feats = LDS row 0 ----
  {
    float s = b1[lane];
    #pragma unroll 8
    for (int c = 0; c < CC; ++c) s = fmaf(F[c], W1[c * HID + lane], s);
    H1[w][lane] = (s >= 0.0f) ? s : SLOPE * s;
  }
  __syncthreads();

  // ---- conv weights: each lane computes 4 of 128 slots (j>=120 -> 0) ----
  #pragma unroll
  for (int i = 0; i < 4; ++i) {
    const int j = lane + 32 * i;
    float s = 0.0f;
    if (j < KGDIM) {
      s = b2[j];
      #pragma unroll 8
      for (int u = 0; u < HID; ++u) s = fmaf(H1[w][u], W2[u * KGDIM + j], s);
    }
    CW[w][j] = s;
  }
  __syncthreads();

  // ---- einsum via V_WMMA_F32_16X16X4_F32 ----
  // A (16x4): rows = kernel point, cols = h-chunk; lanes 0-15 -> K=0/1, 16-31 -> K=2/3
  // B (4x16): rows = h, cols = c-tile; mirrored lane layout
  const int n      = lane & 15;               // row (A) / column (B,D) index
  const int kksel  = (lane < 16) ? 0 : 2;     // which pair of K slots this half-wave holds

  v8f acc[GG];
  #pragma unroll
  for (int t = 0; t < GG; ++t) acc[t] = (v8f){0.f,0.f,0.f,0.f,0.f,0.f,0.f,0.f};

  #pragma unroll
  for (int chunk = 0; chunk < 8; ++chunk) {
    const int h0 = chunk * 4 + kksel;
    v2f a;
    a.x = I[n * ISTRIDE + h0];
    a.y = I[n * ISTRIDE + h0 + 1];
    #pragma unroll
    for (int t = 0; t < GG; ++t) {
      v2f b;
      b.x = F[(h0)     * FSTRIDE + t * 16 + n];
      b.y = F[(h0 + 1) * FSTRIDE + t * 16 + n];
      acc[t] = __builtin_amdgcn_wmma_f32_16x16x4_f32(
          false, a, false, b, (short)0, acc[t], false, false);
    }
  }

  // ---- K-reduction with conv weights ----
  // D layout: VGPR r -> k=r (lanes 0-15), k=r+8 (lanes 16-31); column = lane&15
  const int kbase = (lane < 16) ? 0 : 8;
  #pragma unroll
  for (int t = 0; t < GG; ++t) {
    float p = 0.0f;
    #pragma unroll
    for (int r = 0; r < 8; ++r)
      p = fmaf(acc[t][r], CW[w][(kbase + r) * GG + t], p);
    p += __shfl_xor(p, 16, 32);               // combine k=0..7 with k=8..15
    if (act && lane < 16)
      out[(size_t)m * CC + t * 16 + lane] = p;
  }
}

// ---------------------------------------------------------------------------
// Deterministic batch-norm: per-block partial sums -> finalize -> normalize.
// ---------------------------------------------------------------------------
__global__ __launch_bounds__(CC) void kp_stats_partial(
    const float* __restrict__ out, float* __restrict__ psum,
    float* __restrict__ psq, int M, int rows)
{
  const int c  = threadIdx.x;
  const int r0 = blockIdx.x * rows;
  int r1 = r0 + rows; if (r1 > M) r1 = M;
  float s = 0.0f, sq = 0.0f;
  for (int r = r0; r < r1; ++r) {
    const float v = out[(size_t)r * CC + c];
    s += v;
    sq = fmaf(v, v, sq);
  }
  psum[blockIdx.x * CC + c] = s;
  psq [blockIdx.x * CC + c] = sq;
}

__global__ __launch_bounds__(CC) void kp_stats_final(
    const float* __restrict__ psum, const float* __restrict__ psq,
    float* __restrict__ meanv, float* __restrict__ invstd, int nb, float invM)
{
  const int c = threadIdx.x;
  float s = 0.0f, sq = 0.0f;
  for (int b = 0; b < nb; ++b) { s += psum[b * CC + c]; sq += psq[b * CC + c]; }
  const float mean = s * invM;
  const float var  = fmaf(-mean, mean, sq * invM);
  meanv[c]  = mean;
  invstd[c] = rsqrtf(var + 1e-5f);
}

__global__ __launch_bounds__(256) void kp_normalize(
    float* __restrict__ out, const float* __restrict__ meanv,
    const float* __restrict__ invstd, const float* __restrict__ gamma,
    const float* __restrict__ beta, int total)
{
  const int i = blockIdx.x * 256 + threadIdx.x;
  if (i >= total) return;
  const int c = i & (CC - 1);
  float v = (out[i] - meanv[c]) * invstd[c] * gamma[c] + beta[c];
  out[i] = (v >= 0.0f) ? v : SLOPE * v;
}

// ---------------------------------------------------------------------------
extern "C" void kernel_launch(void* const* d_in, const int* in_sizes, int n_in,
                              void* d_out, int out_size, void* d_ws, size_t ws_size,
                              hipStream_t stream) {
  const float* q_pts   = (const float*)d_in[0];
  const float* s_pts   = (const float*)d_in[1];
  const float* s_feats = (const float*)d_in[2];
  const int*   nbr     = (const int*)  d_in[3];
  const float* kpts    = (const float*)d_in[4];
  const float* W1      = (const float*)d_in[5];
  const float* b1      = (const float*)d_in[6];
  const float* W2      = (const float*)d_in[7];
  const float* b2      = (const float*)d_in[8];
  const float* gamma   = (const float*)d_in[9];
  const float* beta    = (const float*)d_in[10];

  const int M = in_sizes[0] / 3;
  float* out = (float*)d_out;

  // workspace: psum[nb*128], psq[nb*128], mean[128], invstd[128]
  const int ROWS = 256;
  const int nb = (M + ROWS - 1) / ROWS;
  float* psum   = (float*)d_ws;
  float* psq    = psum + (size_t)nb * CC;
  float* meanv  = psq  + (size_t)nb * CC;
  float* invstd = meanv + CC;

  const int mainBlocks = (M + WAVES - 1) / WAVES;
  kp_main<<<mainBlocks, WAVES * 32, 0, stream>>>(
      q_pts, s_pts, s_feats, nbr, kpts, W1, b1, W2, b2, out, M);

  kp_stats_partial<<<nb, CC, 0, stream>>>(out, psum, psq, M, ROWS);
  kp_stats_final<<<1, CC, 0, stream>>>(psum, psq, meanv, invstd, nb, 1.0f / (float)M);

  const int total = M * CC;
  kp_normalize<<<(total + 255) / 256, 256, 0, stream>>>(out, meanv, invstd, gamma, beta, total);
}